// GNNModel_46471546143561
// MI455X (gfx1250) — compile-verified
//
#include <hip/hip_runtime.h>
#include <hip/hip_bf16.h>

typedef __attribute__((ext_vector_type(2))) float v2f;
typedef __attribute__((ext_vector_type(8))) float v8f;

#define N_FEAT 128

// ---------------- utility kernels ----------------

__global__ void zero_f32_kernel(float* __restrict__ p, long n) {
  long i = (long)blockIdx.x * blockDim.x + threadIdx.x;
  long stride = (long)gridDim.x * blockDim.x;
  for (; i < n; i += stride) p[i] = 0.0f;
}

__global__ void deg_count_kernel(const int* __restrict__ dst, int E, int* __restrict__ deg) {
  int e = blockIdx.x * blockDim.x + threadIdx.x;
  if (e < E) atomicAdd(&deg[dst[e]], 1);
}

__global__ void dinv_kernel(const int* __restrict__ deg, float* __restrict__ dinv, int n) {
  int i = blockIdx.x * blockDim.x + threadIdx.x;
  if (i < n) dinv[i] = rsqrtf((float)deg[i] + 1.0f);
}

// ---------------- fp32 WMMA GEMM:  C[M,NCOLS] = A[M,128] @ B[128,NCOLS] ----------------
// One wave computes a 16x64 strip (4 accumulators, A fragment reused 4x per K-step).
// V_WMMA_F32_16X16X4_F32: A 16x4 (v2f/lane), B 4x16 (v2f/lane), C/D 16x16 (v8f/lane).

template <int NCOLS>
__global__ void gemm_wmma_kernel(const float* __restrict__ A,
                                 const float* __restrict__ B,
                                 float* __restrict__ C,
                                 int M) {
  constexpr int NGRP = NCOLS / 64;
  const int lane = threadIdx.x & 31;
  const int wave = threadIdx.x >> 5;
  const int w = blockIdx.x * (blockDim.x >> 5) + wave;
  const int mtile = w / NGRP;
  const int ng = w % NGRP;
  if (mtile * 16 >= M) return;  // wave-uniform: EXEC stays all-ones for live waves

  const int half = lane >> 4;   // 0: lanes 0-15, 1: lanes 16-31
  const int l16  = lane & 15;
  const long arow  = (long)(mtile * 16 + l16) * N_FEAT;
  const int  cbase = ng * 64;

  v8f acc[4] = {};
  for (int k = 0; k < N_FEAT; k += 4) {
    const int ka = k + 2 * half;
    v2f a;
    a.x = A[arow + ka];
    a.y = A[arow + ka + 1];
#pragma unroll
    for (int t = 0; t < 4; ++t) {
      const int col = cbase + t * 16 + l16;
      v2f b;
      b.x = B[(long)ka * NCOLS + col];
      b.y = B[(long)(ka + 1) * NCOLS + col];
      acc[t] = __builtin_amdgcn_wmma_f32_16x16x4_f32(
          /*neg_a=*/false, a, /*neg_b=*/false, b,
          /*c_mod=*/(short)0, acc[t], /*reuse_a=*/false, /*reuse_b=*/false);
    }
  }

#pragma unroll
  for (int t = 0; t < 4; ++t) {
#pragma unroll
    for (int r = 0; r < 8; ++r) {
      const int row = mtile * 16 + r + 8 * half;
      const int col = cbase + t * 16 + l16;
      C[(long)row * NCOLS + col] = acc[t][r];
    }
  }
}

// ---------------- edge scatter: agg[dst] += h[src] * dinv[src]*dinv[dst] ----------------

template <int NCOLS>
__global__ void scatter_kernel(const int* __restrict__ src, const int* __restrict__ dst,
                               const float* __restrict__ dinv, const float* __restrict__ h,
                               float* __restrict__ agg, int E) {
  const int e = blockIdx.x;
  const int s = src[e];
  const int d = dst[e];
  const float norm = dinv[s] * dinv[d];
  const float* hs = h + (long)s * NCOLS;
  float* ad = agg + (long)d * NCOLS;
  for (int c = threadIdx.x; c < NCOLS; c += blockDim.x)
    unsafeAtomicAdd(&ad[c], hs[c] * norm);  // native global_atomic_add_f32
}

// ---------------- finalize: self-loop + bias (+ ReLU for layer 1) ----------------

__global__ void finalize_relu_kernel(const float* __restrict__ agg, const float* __restrict__ h,
                                     const float* __restrict__ dinv, const float* __restrict__ bias,
                                     float* __restrict__ out, int total) {
  int i = blockIdx.x * blockDim.x + threadIdx.x;
  if (i >= total) return;
  int row = i >> 7;   // NCOLS = 128
  int c   = i & 127;
  float di = dinv[row];
  float v  = agg[i] + h[i] * di * di + bias[c];
  out[i] = fmaxf(v, 0.0f);
}

__global__ void finalize2_kernel(const float* __restrict__ agg, const float* __restrict__ dinv,
                                 const float* __restrict__ bias, float* __restrict__ emb,
                                 int total) {
  int i = blockIdx.x * blockDim.x + threadIdx.x;
  if (i >= total) return;
  int row = i >> 9;   // NCOLS = 512
  int c   = i & 511;
  float di = dinv[row];
  emb[i] = agg[i] + emb[i] * di * di + bias[c];  // emb currently holds h2
}

// ---------------- query edges: 8 lanes per edge, one head per lane ----------------

__global__ void query_kernel(const int* __restrict__ qrow, const int* __restrict__ qcol,
                             const int* __restrict__ y, const float* __restrict__ emb,
                             float* __restrict__ pred_i, float* __restrict__ pred_j,
                             float* __restrict__ pred, int Q) {
  const int gid = blockIdx.x * blockDim.x + threadIdx.x;
  const int e = gid >> 3;
  const int h = gid & 7;
  if (e >= Q) return;
  const int r = qrow[e];
  const int c = qcol[e];
  const float4* er = (const float4*)(emb + (long)r * 512 + h * 64);
  const float4* ec = (const float4*)(emb + (long)c * 512 + h * 64);
  float s = 0.0f;
#pragma unroll
  for (int i = 0; i < 16; ++i) {
    float4 a = er[i];
    float4 b = ec[i];
    s += a.x * b.x + a.y * b.y + a.z * b.z + a.w * b.w;
  }
  const int lane = threadIdx.x & 31;
  const int base = lane & ~7;           // first lane of this edge's 8-lane group
  const float pi = __shfl(s, base + y[r], 32);
  const float pj = __shfl(s, base + y[c], 32);
  if (h == 0) {
    pred_i[e] = pi;
    pred_j[e] = pj;
    pred[e]   = 0.5f * (pi + pj);
  }
}

// ---------------- launcher ----------------

extern "C" void kernel_launch(void* const* d_in, const int* in_sizes, int n_in,
                              void* d_out, int out_size, void* d_ws, size_t ws_size,
                              hipStream_t stream) {
  const float* x  = (const float*)d_in[0];
  const float* W1 = (const float*)d_in[1];
  const float* b1 = (const float*)d_in[2];
  const float* W2 = (const float*)d_in[3];
  const float* b2 = (const float*)d_in[4];
  const int* edge_index   = (const int*)d_in[5];
  const int* q_edge_index = (const int*)d_in[6];
  const int* y            = (const int*)d_in[7];

  const int N = in_sizes[0] / 128;
  const int E = in_sizes[5] / 2;
  const int Q = in_sizes[6] / 2;

  const int* src  = edge_index;
  const int* dst  = edge_index + E;
  const int* qrow = q_edge_index;
  const int* qcol = q_edge_index + Q;

  // output layout: hiddens [N,128] | emb [N,512] | pred_i [Q] | pred_j [Q] | pred [Q]
  float* out     = (float*)d_out;
  float* hiddens = out;
  float* emb     = out + (long)N * 128;
  float* pred_i  = emb + (long)N * 512;
  float* pred_j  = pred_i + Q;
  float* pred    = pred_j + Q;

  // workspace: dinv | deg(int) | h1 [N,128] | agg [N,512] (reused agg1 -> agg2)
  const long npad = (long)((N + 1023) & ~1023);
  float* wsf  = (float*)d_ws;
  float* dinv = wsf;
  int*   deg  = (int*)(wsf + npad);
  float* h1   = wsf + 2 * npad;
  float* agg  = h1 + (long)N * 128;

  // ---- degrees / normalization ----
  zero_f32_kernel<<<512, 256, 0, stream>>>((float*)deg, (long)N);
  deg_count_kernel<<<(E + 255) / 256, 256, 0, stream>>>(dst, E, deg);
  dinv_kernel<<<(N + 255) / 256, 256, 0, stream>>>(deg, dinv, N);

  // ---- layer 1: h1 = x @ W1 ----
  {
    const int waves = ((N + 15) / 16) * (128 / 64);
    gemm_wmma_kernel<128><<<(waves + 7) / 8, 256, 0, stream>>>(x, W1, h1, N);
  }
  zero_f32_kernel<<<2048, 256, 0, stream>>>(agg, (long)N * 128);
  scatter_kernel<128><<<E, 128, 0, stream>>>(src, dst, dinv, h1, agg, E);
  finalize_relu_kernel<<<((N * 128) + 255) / 256, 256, 0, stream>>>(
      agg, h1, dinv, b1, hiddens, N * 128);

  // ---- layer 2: h2 (stored in emb slot) = hiddens @ W2 ----
  {
    const int waves = ((N + 15) / 16) * (512 / 64);
    gemm_wmma_kernel<512><<<(waves + 7) / 8, 256, 0, stream>>>(hiddens, W2, emb, N);
  }
  zero_f32_kernel<<<4096, 256, 0, stream>>>(agg, (long)N * 512);
  scatter_kernel<512><<<E, 256, 0, stream>>>(src, dst, dinv, emb, agg, E);
  finalize2_kernel<<<((N * 512) + 255) / 256, 256, 0, stream>>>(
      agg, dinv, b2, emb, N * 512);

  // ---- query edges ----
  query_kernel<<<((Q * 8) + 255) / 256, 256, 0, stream>>>(
      qrow, qcol, y, emb, pred_i, pred_j, pred, Q);
}